// FastNetwork_65395172049296
// MI455X (gfx1250) — compile-verified
//
#include <hip/hip_runtime.h>
#include <stdint.h>

// ---------------------------------------------------------------------------
// Sequential fast-weight scan, parallelized exactly via monotone bracketing.
//   s <- sigmoid(10*(s + u_i - 0.5)) ; pred_i = x_i . s
// Chunked: 32768 chunks x 128 steps, lane-per-chunk (wave32), TDM tile loads.
// ---------------------------------------------------------------------------

typedef unsigned int u32;
typedef unsigned int u32x4 __attribute__((ext_vector_type(4)));
typedef unsigned int u32x8 __attribute__((ext_vector_type(8)));

constexpr int kN      = 4194304;
constexpr int kChunks = 32768;
constexpr int kL      = kN / kChunks;          // 128 steps per chunk
constexpr int kWaves  = 4;                     // waves per block
constexpr int kBlock  = kWaves * 32;           // 128 threads
constexpr int kCPB    = kWaves * 32;           // chunks per block = 128
constexpr int kGrid   = kChunks / kCPB;        // 256 blocks

// Pass-1 tiling (update only): 32 steps/tile
constexpr int kT1     = 32;
constexpr int kRow1   = 3 * kT1;               // 96 dwords per chunk-row
constexpr int kPitch1 = kRow1 + kRow1 / 16;    // 102 dwords (TDM pad 1dw/16dw)
constexpr int kWBuf1  = 32 * kPitch1;          // dwords per buffer per wave
// Pass-3 tiling (update + x): 16 steps/tile
constexpr int kT3     = 16;
constexpr int kRow3   = 3 * kT3;               // 48 dwords
constexpr int kPitch3 = kRow3 + kRow3 / 16;    // 51 dwords (odd -> bank-conflict-free)
constexpr int kWBuf3  = 32 * kPitch3;

// Both passes use 104448 B of dynamic LDS per 128-thread block (<< 320KB/WGP):
//   pass1: 4 waves * 2 bufs * kWBuf1*4  = 104448
//   pass3: 4 waves * 2 bufs * 2 arrays * kWBuf3*4 = 104448
constexpr int kLdsBytes = kWaves * 2 * kWBuf1 * 4;
static_assert(kWaves * 2 * kWBuf1 * 4 == kWaves * 4 * kWBuf3 * 4, "lds sizes match");

// sigmoid(10*(s+u-0.5)) = 1 / (1 + exp2( A*s + (A*u + B) )),  A=-10*log2(e)
#define SIG_A (-14.426950408889634f)
#define SIG_B (7.213475204444817f)
#define PADIDX(w) ((w) + ((w) >> 4))   // TDM inserted 1 pad dword every 16 dwords

__device__ __forceinline__ float sig_step(float s, float cu) {
  float z = __builtin_fmaf(s, SIG_A, cu);
  float e = __builtin_amdgcn_exp2f(z);          // v_exp_f32
  return __builtin_amdgcn_rcpf(1.0f + e);       // v_rcp_f32
}

// --- Tensor Data Mover: 2D tile (32 rows x row_dw dwords, stride stride_dw)
// from global -> LDS at byte offset lds_off. Pad: +1 dword every 16 dwords so
// per-lane LDS rows hit distinct banks. D# bitfields per CDNA5 ISA §8.3/8.4.
__device__ __forceinline__ void tdm_load_2d(u32 lds_off, const float* gptr,
                                            u32 row_dw, u32 stride_dw) {
  const unsigned long long ga = (unsigned long long)(uintptr_t)gptr;
  u32x4 g0;
  g0[0] = 1u;                                   // count=1 valid descriptor
  g0[1] = lds_off;                              // lds_addr (bytes)
  g0[2] = (u32)ga;                              // global_addr[31:0]
  g0[3] = (u32)(ga >> 32) | (2u << 30);         // global_addr[56:32] | type=2
  u32x8 g1;
  g1[0] = (2u << 16)                            // data_size = 4 bytes
        | (1u << 20)                            // pad_enable
        | (3u << 22);                           // pad_interval: every 16 dwords
                                                // pad_amount field 0 => 1 dword
  g1[1] = (row_dw & 0xFFFFu) << 16;             // tensor_dim0[15:0]
  g1[2] = (row_dw >> 16) | (32u << 16);         // tensor_dim0[31:16] | tensor_dim1
  g1[3] = (row_dw & 0xFFFFu) << 16;             // tensor_dim1[31:16]=0 | tile_dim0
  g1[4] = 32u;                                  // tile_dim1 = 32 rows (tile_dim2=0)
  g1[5] = stride_dw;                            // tensor_dim0_stride[31:0]
  g1[6] = 0u;                                   // stride[47:32] | dim1_stride lo
  g1[7] = 0u;
  asm volatile("tensor_load_to_lds %0, %1" :: "s"(g0), "s"(g1) : "memory");
}

__device__ __forceinline__ void wait_tensorcnt_0() {
  asm volatile("s_wait_tensorcnt 0" ::: "memory");
}
__device__ __forceinline__ void wait_tensorcnt_1() {
  asm volatile("s_wait_tensorcnt 1" ::: "memory");
}
__device__ __forceinline__ void wait_tensorcnt_2() {
  asm volatile("s_wait_tensorcnt 2" ::: "memory");
}

// ---------------------------------------------------------------------------
// Pass 1: per-chunk bracket run. Lane = chunk. lower starts at 0, upper at 1
// (monotone map => brackets any incoming state in [0,1]). Chunk 0 runs the
// true initial state so its output is exact unconditionally.
// ---------------------------------------------------------------------------
__global__ __launch_bounds__(kBlock)
void pass1_bounds(const float* __restrict__ upd, const float* __restrict__ net0,
                  float* __restrict__ loB, float* __restrict__ hiB) {
  extern __shared__ float smem[];
  const int lane = threadIdx.x & 31;
  const int wid  = __builtin_amdgcn_readfirstlane((int)(threadIdx.x >> 5));
  const int wc0  = blockIdx.x * kCPB + wid * 32;     // wave's first chunk
  const int c    = wc0 + lane;

  float l0 = 0.0f, l1 = 0.0f, l2 = 0.0f;
  float h0 = 1.0f, h1 = 1.0f, h2 = 1.0f;
  if (c == 0) { l0 = h0 = net0[0]; l1 = h1 = net0[1]; l2 = h2 = net0[2]; }

  const u32 waveLds = (u32)(wid * 2 * kWBuf1 * 4);
  const float* gbase = upd + (size_t)wc0 * kL * 3;
  constexpr int nT = kL / kT1;                       // 4 tiles

  tdm_load_2d(waveLds, gbase, (u32)kRow1, (u32)(3 * kL));
  for (int tile = 0; tile < nT; ++tile) {
    if (tile + 1 < nT) {
      tdm_load_2d(waveLds + (u32)(((tile + 1) & 1) * kWBuf1 * 4),
                  gbase + (size_t)(tile + 1) * kT1 * 3, (u32)kRow1, (u32)(3 * kL));
      wait_tensorcnt_1();                            // current tile resident
    } else {
      wait_tensorcnt_0();
    }
    const float* buf = smem + (waveLds >> 2) + (tile & 1) * kWBuf1 + lane * kPitch1;
#pragma unroll
    for (int t = 0; t < kT1; ++t) {
      const int w = 3 * t;
      const float u0 = buf[PADIDX(w)];
      const float u1 = buf[PADIDX(w + 1)];
      const float u2 = buf[PADIDX(w + 2)];
      const float c0 = __builtin_fmaf(u0, SIG_A, SIG_B);
      const float c1 = __builtin_fmaf(u1, SIG_A, SIG_B);
      const float c2 = __builtin_fmaf(u2, SIG_A, SIG_B);
      l0 = sig_step(l0, c0); h0 = sig_step(h0, c0);
      l1 = sig_step(l1, c1); h1 = sig_step(h1, c1);
      l2 = sig_step(l2, c2); h2 = sig_step(h2, c2);
    }
  }
  loB[c * 3 + 0] = l0; loB[c * 3 + 1] = l1; loB[c * 3 + 2] = l2;
  hiB[c * 3 + 0] = h0; hiB[c * 3 + 1] = h1; hiB[c * 3 + 2] = h2;
}

// ---------------------------------------------------------------------------
// Pass 2a: boundary-in state of chunk j = collapsed output of chunk j-1.
// Non-collapsed predecessors are appended to a fixup list.
// ---------------------------------------------------------------------------
__global__ __launch_bounds__(256)
void pass2a(const float* __restrict__ loB, const float* __restrict__ hiB,
            const float* __restrict__ net0, float* __restrict__ bin,
            u32* __restrict__ fixCnt, u32* __restrict__ fixList) {
  const int j = blockIdx.x * 256 + threadIdx.x;
  if (j >= kChunks) return;
  if (j == 0) {
    bin[0] = net0[0]; bin[1] = net0[1]; bin[2] = net0[2];
    return;
  }
  const float a0 = loB[(j - 1) * 3 + 0], a1 = loB[(j - 1) * 3 + 1], a2 = loB[(j - 1) * 3 + 2];
  const float b0 = hiB[(j - 1) * 3 + 0], b1 = hiB[(j - 1) * 3 + 1], b2 = hiB[(j - 1) * 3 + 2];
  bin[j * 3 + 0] = a0; bin[j * 3 + 1] = a1; bin[j * 3 + 2] = a2;
  if (a0 != b0 || a1 != b1 || a2 != b2) {
    const u32 idx = atomicAdd(fixCnt, 1u);
    if (idx < (u32)kChunks) fixList[idx] = (u32)j;   // bin[j] is suspect
  }
}

// ---------------------------------------------------------------------------
// Pass 2b: sequential exact fixup of the (expected-empty) suspect set, in
// ascending chunk order so corrected states propagate.
// ---------------------------------------------------------------------------
__global__ void pass2b(const float* __restrict__ upd, float* __restrict__ bin,
                       const u32* __restrict__ fixCnt, const u32* __restrict__ fixList) {
  if (threadIdx.x != 0 || blockIdx.x != 0) return;
  u32 n = *fixCnt; if (n > (u32)kChunks) n = (u32)kChunks;
  u32 last = 0;
  for (u32 done = 0; done < n; ++done) {
    u32 best = 0xFFFFFFFFu;                          // selection: ascending order
    for (u32 i = 0; i < n; ++i) {
      const u32 v = fixList[i];
      if (v >= last && v < best) best = v;
    }
    if (best == 0xFFFFFFFFu) break;
    last = best + 1;
    const int j = (int)best;                         // recompute chunk j-1 exactly
    float s0 = bin[(j - 1) * 3 + 0], s1 = bin[(j - 1) * 3 + 1], s2 = bin[(j - 1) * 3 + 2];
    const float* up = upd + (size_t)(j - 1) * kL * 3;
    for (int t = 0; t < kL; ++t) {
      s0 = sig_step(s0, __builtin_fmaf(up[3 * t + 0], SIG_A, SIG_B));
      s1 = sig_step(s1, __builtin_fmaf(up[3 * t + 1], SIG_A, SIG_B));
      s2 = sig_step(s2, __builtin_fmaf(up[3 * t + 2], SIG_A, SIG_B));
    }
    bin[j * 3 + 0] = s0; bin[j * 3 + 1] = s1; bin[j * 3 + 2] = s2;
  }
}

// ---------------------------------------------------------------------------
// Pass 3: re-run each chunk from its exact boundary state; stream update AND x
// via TDM (double-buffered pairs, wait tensorcnt<=2), emit predictions.
// ---------------------------------------------------------------------------
__global__ __launch_bounds__(kBlock)
void pass3_preds(const float* __restrict__ x, const float* __restrict__ upd,
                 const float* __restrict__ bin, float* __restrict__ preds) {
  extern __shared__ float smem[];
  const int lane = threadIdx.x & 31;
  const int wid  = __builtin_amdgcn_readfirstlane((int)(threadIdx.x >> 5));
  const int wc0  = blockIdx.x * kCPB + wid * 32;
  const int c    = wc0 + lane;

  float s0 = bin[c * 3 + 0], s1 = bin[c * 3 + 1], s2 = bin[c * 3 + 2];

  // per-wave LDS: [buf0_u][buf0_x][buf1_u][buf1_x], each kWBuf3 dwords
  const u32 waveLds = (u32)(wid * 4 * kWBuf3 * 4);
  const size_t gofs = (size_t)wc0 * kL * 3;
  constexpr int nT = kL / kT3;                       // 8 tiles
  float* outp = preds + (size_t)c * kL;

  tdm_load_2d(waveLds,                 upd + gofs, (u32)kRow3, (u32)(3 * kL));
  tdm_load_2d(waveLds + (u32)(kWBuf3 * 4), x + gofs, (u32)kRow3, (u32)(3 * kL));
  for (int tile = 0; tile < nT; ++tile) {
    if (tile + 1 < nT) {
      const u32 off = waveLds + (u32)(((tile + 1) & 1) * 2 * kWBuf3 * 4);
      const size_t g = gofs + (size_t)(tile + 1) * kT3 * 3;
      tdm_load_2d(off,                     upd + g, (u32)kRow3, (u32)(3 * kL));
      tdm_load_2d(off + (u32)(kWBuf3 * 4), x + g,   (u32)kRow3, (u32)(3 * kL));
      wait_tensorcnt_2();                            // current pair resident
    } else {
      wait_tensorcnt_0();
    }
    const float* bu = smem + (waveLds >> 2) + (tile & 1) * 2 * kWBuf3 + lane * kPitch3;
    const float* bx = bu + kWBuf3;
#pragma unroll
    for (int t = 0; t < kT3; ++t) {
      const int w = 3 * t;
      const float u0 = bu[PADIDX(w)], u1 = bu[PADIDX(w + 1)], u2 = bu[PADIDX(w + 2)];
      const float x0 = bx[PADIDX(w)], x1 = bx[PADIDX(w + 1)], x2 = bx[PADIDX(w + 2)];
      s0 = sig_step(s0, __builtin_fmaf(u0, SIG_A, SIG_B));
      s1 = sig_step(s1, __builtin_fmaf(u1, SIG_A, SIG_B));
      s2 = sig_step(s2, __builtin_fmaf(u2, SIG_A, SIG_B));
      outp[tile * kT3 + t] = __builtin_fmaf(x0, s0, __builtin_fmaf(x1, s1, x2 * s2));
    }
  }
}

// ---------------------------------------------------------------------------
// Host launch. Workspace layout (all within d_ws):
//   loB  : kChunks*3 f32
//   hiB  : kChunks*3 f32
//   bin  : kChunks*3 f32
//   fixCnt : 1 u32  (+pad)
//   fixList: kChunks u32
// Total ~1.31 MB.
// ---------------------------------------------------------------------------
extern "C" void kernel_launch(void* const* d_in, const int* in_sizes, int n_in,
                              void* d_out, int out_size, void* d_ws, size_t ws_size,
                              hipStream_t stream) {
  (void)in_sizes; (void)n_in; (void)out_size; (void)ws_size;
  const float* x    = (const float*)d_in[0];   // [N,1,3]
  const float* upd  = (const float*)d_in[1];   // [N,3,1]
  const float* net0 = (const float*)d_in[2];   // [3,1]
  float* preds = (float*)d_out;                // [N,1]

  char* ws = (char*)d_ws;
  const size_t szB = (size_t)kChunks * 3 * sizeof(float);  // 393216
  float* loB = (float*)(ws);
  float* hiB = (float*)(ws + szB);
  float* bin = (float*)(ws + 2 * szB);
  u32* fixCnt  = (u32*)(ws + 3 * szB);
  u32* fixList = (u32*)(ws + 3 * szB + 16);

  // Allow >64KB dynamic LDS (320KB/WGP on CDNA5). Harmless if it errors.
  hipFuncSetAttribute((const void*)pass1_bounds,
                      hipFuncAttributeMaxDynamicSharedMemorySize, kLdsBytes);
  hipFuncSetAttribute((const void*)pass3_preds,
                      hipFuncAttributeMaxDynamicSharedMemorySize, kLdsBytes);

  hipMemsetAsync(fixCnt, 0, sizeof(u32), stream);
  pass1_bounds<<<kGrid, kBlock, kLdsBytes, stream>>>(upd, net0, loB, hiB);
  pass2a<<<kChunks / 256, 256, 0, stream>>>(loB, hiB, net0, bin, fixCnt, fixList);
  pass2b<<<1, 1, 0, stream>>>(upd, bin, fixCnt, fixList);
  pass3_preds<<<kGrid, kBlock, kLdsBytes, stream>>>(x, upd, bin, preds);
}